// DSSIMLoss_12283606467878
// MI455X (gfx1250) — compile-verified
//
#include <hip/hip_runtime.h>

typedef float v2f __attribute__((ext_vector_type(2)));
typedef float v8f __attribute__((ext_vector_type(8)));
typedef int   v4i __attribute__((ext_vector_type(4)));

#define AS1 __attribute__((address_space(1)))
#define AS3 __attribute__((address_space(3)))

// ---------------- problem constants (fixed by setup_inputs) ----------------
constexpr int W = 512, H = 512, CHN = 3, NIMG = 8;
constexpr int HW = W * H;          // 262144
constexpr int CHW = CHN * HW;      // 786432
constexpr float C1c = 1e-4f;       // (0.01)^2
constexpr float C2c = 9e-4f;       // (0.03)^2

// ---------------- tiling ----------------
constexpr int TW = 256;            // output tile width  (one thread per column)
constexpr int THR = 4;             // output tile height (rows per thread)
constexpr int ROWS = THR + 2;      // 6 staged rows (with vertical halo)
constexpr int LSTRIDE = 264;       // floats per LDS row (col c stored at (c-x0)+4)
constexpr int NBUF = 6;            // 2 images * 3 channels
constexpr int SMEMF = NBUF * ROWS * LSTRIDE;  // 9504 floats = 38016 B

// ---------------- CDNA5 async global->LDS copies ----------------
__device__ __forceinline__ void async_b128(const float* g, float* l) {
#if __has_builtin(__builtin_amdgcn_global_load_async_to_lds_b128)
  __builtin_amdgcn_global_load_async_to_lds_b128((AS1 v4i*)g, (AS3 v4i*)l, 0, 0);
#else
  asm volatile("global_load_async_to_lds_b128 %0, %1, off"
               :: "v"((AS3 float*)l), "v"(g) : "memory");
#endif
}

__device__ __forceinline__ void async_b32(const float* g, float* l) {
#if __has_builtin(__builtin_amdgcn_global_load_async_to_lds_b32)
  __builtin_amdgcn_global_load_async_to_lds_b32((AS1 int*)g, (AS3 int*)l, 0, 0);
#else
  asm volatile("global_load_async_to_lds_b32 %0, %1, off"
               :: "v"((AS3 float*)l), "v"(g) : "memory");
#endif
}

__device__ __forceinline__ void wait_async0() {
#if __has_builtin(__builtin_amdgcn_s_wait_asynccnt)
  __builtin_amdgcn_s_wait_asynccnt(0);
#else
  asm volatile("s_wait_asynccnt 0" ::: "memory");
#endif
}

// ---------------- block reduction: 256 lanes -> 1 (deterministic) ----------
// First step uses V_WMMA_F32_16X16X4_F32 with A=ones: D[m][n] = sum_k B[k][n],
// i.e. the 32 per-lane partials collapse to 16 column sums (each lane of the
// first D register holds columnsum[lane & 15]).
__device__ __forceinline__ float block_sum_256(float v, float* red) {
#if defined(__gfx1250__) && __has_builtin(__builtin_amdgcn_wmma_f32_16x16x4_f32)
  v2f a; a[0] = 1.0f; a[1] = 1.0f;       // A = ones(16x4)
  v2f b; b[0] = v;    b[1] = 0.0f;       // B rows hold the 32 partials
  v8f c = {};
  v8f d = __builtin_amdgcn_wmma_f32_16x16x4_f32(
      /*neg_a=*/false, a, /*neg_b=*/false, b,
      /*c_mod=*/(short)0, c, /*reuse_a=*/false, /*reuse_b=*/false);
  const int lane = (int)(threadIdx.x & 31u);
  const int wave = (int)(threadIdx.x >> 5);
  if (lane < 16) red[wave * 16 + lane] = d[0];  // 16 distinct sums per wave
  __syncthreads();
  for (int s = 64; s > 0; s >>= 1) {            // 128 values -> 1
    if ((int)threadIdx.x < s) red[threadIdx.x] += red[threadIdx.x + s];
    __syncthreads();
  }
#else
  red[threadIdx.x] = v;
  __syncthreads();
  for (int s = 128; s > 0; s >>= 1) {
    if ((int)threadIdx.x < s) red[threadIdx.x] += red[threadIdx.x + s];
    __syncthreads();
  }
#endif
  return red[0];
}

// ---------------- kernel 1: per-tile DSSIM partial sums ----------------
__global__ __launch_bounds__(256) void dssim_partial(const float* __restrict__ yp,
                                                     const float* __restrict__ yt,
                                                     float* __restrict__ part) {
  __shared__ __align__(16) float sm[SMEMF];
  __shared__ float red[256];

  const int tid = (int)threadIdx.x;
  const int x0 = (int)blockIdx.x * TW;
  const int y0 = (int)blockIdx.y * THR;
  const int n  = (int)blockIdx.z;

  // 1) zero LDS (realizes the zero padding for image borders / skipped rows)
  float4* sm4 = (float4*)sm;
  const float4 z4 = make_float4(0.f, 0.f, 0.f, 0.f);
  for (int i = tid; i < SMEMF / 4; i += 256) sm4[i] = z4;
  __syncthreads();

  // 2) async-stage tile+halo into LDS.
  // Body: 6 buffers * 6 rows * 64 aligned 16B chunks = 2304 = 256 threads * 9.
#pragma unroll
  for (int i = 0; i < 9; ++i) {
    const int q = tid + i * 256;
    const int chunk = q & 63;
    const int r = (q >> 6) % ROWS;
    const int buf = q / (64 * ROWS);
    const int img = buf / 3, ch = buf % 3;
    const int gy = y0 - 1 + r;
    if (gy >= 0 && gy < H) {
      const float* src = (img == 0) ? yp : yt;
      const float* g = src + (size_t)n * CHW + (size_t)ch * HW +
                       (size_t)gy * W + (size_t)(x0 + 4 * chunk);
      float* l = &sm[buf * (ROWS * LSTRIDE) + r * LSTRIDE + 4 + 4 * chunk];
      async_b128(g, l);
    }
  }
  // Halo columns: 6 buffers * 6 rows * 2 sides = 72 single-float copies.
  if (tid < NBUF * ROWS * 2) {
    const int side = tid & 1;
    const int r = (tid >> 1) % ROWS;
    const int buf = tid / (ROWS * 2);
    const int img = buf / 3, ch = buf % 3;
    const int gy = y0 - 1 + r;
    const int gx = side ? (x0 + TW) : (x0 - 1);
    if (gy >= 0 && gy < H && gx >= 0 && gx < W) {
      const float* src = (img == 0) ? yp : yt;
      const float* g = src + (size_t)n * CHW + (size_t)ch * HW +
                       (size_t)gy * W + (size_t)gx;
      float* l = &sm[buf * (ROWS * LSTRIDE) + r * LSTRIDE + (side ? (4 + TW) : 3)];
      async_b32(g, l);
    }
  }
  wait_async0();
  __syncthreads();

  // 3) compute: thread owns column x = x0 + tid (LDS col index cb).
  const int cb = tid + 4;
  float Hst[ROWS], Hsp[ROWS], Htt[ROWS], Hpp[ROWS], Htp[ROWS];
#pragma unroll
  for (int r = 0; r < ROWS; ++r) {
    float st = 0.f, sp = 0.f, stt = 0.f, spp = 0.f, stp = 0.f;
#pragma unroll
    for (int dx = 0; dx < 3; ++dx) {
#pragma unroll
      for (int c = 0; c < 3; ++c) {
        const float p = sm[c * (ROWS * LSTRIDE) + r * LSTRIDE + cb - 1 + dx];
        const float t = sm[(3 + c) * (ROWS * LSTRIDE) + r * LSTRIDE + cb - 1 + dx];
        st += t;  sp += p;
        stt = fmaf(t, t, stt);
        spp = fmaf(p, p, spp);
        stp = fmaf(t, p, stp);
      }
    }
    Hst[r] = st; Hsp[r] = sp; Htt[r] = stt; Hpp[r] = spp; Htp[r] = stp;
  }

  float acc = 0.f;
  constexpr float inv27 = 1.0f / 27.0f;
  constexpr float inv26 = 1.0f / 26.0f;
#pragma unroll
  for (int j = 0; j < THR; ++j) {
    const float St  = Hst[j] + Hst[j + 1] + Hst[j + 2];
    const float Sp  = Hsp[j] + Hsp[j + 1] + Hsp[j + 2];
    const float Stt = Htt[j] + Htt[j + 1] + Htt[j + 2];
    const float Spp = Hpp[j] + Hpp[j + 1] + Hpp[j + 2];
    const float Stp = Htp[j] + Htp[j + 1] + Htp[j + 2];
    const float ut = St * inv27, up = Sp * inv27;
    const float vt = (Stt - St * St * inv27) * inv26;   // ddof=1
    const float vp = (Spp - Sp * Sp * inv27) * inv26;
    const float cov = Stp * inv27 - ut * up;
    const float num = (2.f * ut * up + C1c) * (2.f * cov + C2c);
    const float den = (ut * ut + up * up + C1c) * (vp + vt + C2c);
    acc += (1.f - num / den) * 0.5f;
  }

  const float bs = block_sum_256(acc, red);
  if (tid == 0) {
    const int bid = ((int)blockIdx.z * (int)gridDim.y + (int)blockIdx.y) *
                        (int)gridDim.x + (int)blockIdx.x;
    part[bid] = bs;
  }
}

// ---------------- kernel 2: final deterministic reduction ----------------
__global__ __launch_bounds__(256) void dssim_final(const float* __restrict__ part,
                                                   int nparts,
                                                   float* __restrict__ out,
                                                   float inv_count) {
  __shared__ float red[256];
  float acc = 0.f;
  for (int i = (int)threadIdx.x; i < nparts; i += 256) acc += part[i];
  const float total = block_sum_256(acc, red);
  if (threadIdx.x == 0) out[0] = total * inv_count;
}

// ---------------- launch ----------------
extern "C" void kernel_launch(void* const* d_in, const int* in_sizes, int n_in,
                              void* d_out, int out_size, void* d_ws, size_t ws_size,
                              hipStream_t stream) {
  (void)in_sizes; (void)n_in; (void)out_size; (void)ws_size;
  const float* yp = (const float*)d_in[0];   // y_pred [8,3,512,512] f32
  const float* yt = (const float*)d_in[1];   // y_true [8,3,512,512] f32
  float* out = (float*)d_out;                // scalar f32
  float* part = (float*)d_ws;                // 2048 block partials

  dim3 grid(W / TW, H / THR, NIMG);          // (2, 128, 8) = 2048 blocks
  dssim_partial<<<grid, 256, 0, stream>>>(yp, yt, part);

  const int nparts = (W / TW) * (H / THR) * NIMG;
  dssim_final<<<1, 256, 0, stream>>>(part, nparts, out,
                                     1.0f / (float)((size_t)NIMG * H * W));
}